// GATConv_61847529063051
// MI455X (gfx1250) — compile-verified
//
#include <hip/hip_runtime.h>

typedef float v2f __attribute__((ext_vector_type(2)));
typedef float v8f __attribute__((ext_vector_type(8)));

#define NNODES   100000
#define NEDGES   1600000
#define DFEAT    128
#define NEG_SLOPE 0.2f

// ---------------------------------------------------------------- zero fill
__global__ void gat_zero_f32(float* __restrict__ p, long n) {
    long i = blockIdx.x * (long)blockDim.x + threadIdx.x;
    long stride = gridDim.x * (long)blockDim.x;
    for (; i < n; i += stride) p[i] = 0.0f;
}

// ---------------------------------------------------------------- GEMM: feat = h @ W
// One wave computes a 16-row x 128-col strip using V_WMMA_F32_16X16X4_F32.
// A(16x4): lane L holds (M = L%16, K = kbase + (L>=16?2:0) + {0,1}) in 2 VGPRs.
// B(4x16): lane L holds (K = kbase + (L>=16?2:0) + {0,1}, N = L%16) in 2 VGPRs.
// D(16x16): VGPR v <-> row (v + (L>=16?8:0)), col = L%16.
__global__ void gat_gemm_wmma(const float* __restrict__ h,
                              const float* __restrict__ W,
                              float* __restrict__ feat) {
    const int strip = blockIdx.x * 4 + (threadIdx.x >> 5);   // wave-uniform
    if (strip >= NNODES / 16) return;                        // whole wave exits
    const int lane  = threadIdx.x & 31;
    const int m     = lane & 15;
    const int khalf = (lane >> 4) << 1;                      // 0 or 2
    const int hihalf = (lane >> 4) << 3;                     // 0 or 8 (D rows)
    const long row  = (long)strip * 16 + m;

    v8f acc[8];
#pragma unroll
    for (int ct = 0; ct < 8; ++ct) acc[ct] = (v8f){0,0,0,0,0,0,0,0};

    for (int k = 0; k < DFEAT; k += 4) {
        v2f a;
        a.x = h[row * DFEAT + k + khalf + 0];
        a.y = h[row * DFEAT + k + khalf + 1];
#pragma unroll
        for (int ct = 0; ct < 8; ++ct) {
            v2f b;
            b.x = W[(k + khalf + 0) * DFEAT + ct * 16 + m];
            b.y = W[(k + khalf + 1) * DFEAT + ct * 16 + m];
            acc[ct] = __builtin_amdgcn_wmma_f32_16x16x4_f32(
                false, a, false, b, (short)0, acc[ct], false, false);
        }
    }

#pragma unroll
    for (int ct = 0; ct < 8; ++ct) {
#pragma unroll
        for (int v = 0; v < 8; ++v) {
            long orow = (long)strip * 16 + v + hihalf;
            feat[orow * DFEAT + ct * 16 + m] = acc[ct][v];
        }
    }
}

// ---------------------------------------------------------------- el/er = feat @ attn_{l,r}
__global__ void gat_attn_logits(const float* __restrict__ feat,
                                const float* __restrict__ attn_l,
                                const float* __restrict__ attn_r,
                                float* __restrict__ el,
                                float* __restrict__ er) {
    const int node = blockIdx.x * 4 + (threadIdx.x >> 5);
    if (node >= NNODES) return;
    const int lane = threadIdx.x & 31;

    const float4 f  = *(const float4*)(feat   + (long)node * DFEAT + lane * 4);
    const float4 al = *(const float4*)(attn_l + lane * 4);
    const float4 ar = *(const float4*)(attn_r + lane * 4);

    float pl = f.x * al.x + f.y * al.y + f.z * al.z + f.w * al.w;
    float pr = f.x * ar.x + f.y * ar.y + f.z * ar.z + f.w * ar.w;
#pragma unroll
    for (int off = 16; off > 0; off >>= 1) {
        pl += __shfl_xor(pl, off, 32);
        pr += __shfl_xor(pr, off, 32);
    }
    if (lane == 0) { el[node] = pl; er[node] = pr; }
}

// order-preserving float <-> uint encoding for atomic max
__device__ __forceinline__ unsigned enc_f32(float x) {
    unsigned u = __float_as_uint(x);
    return (u & 0x80000000u) ? ~u : (u | 0x80000000u);
}
__device__ __forceinline__ float dec_f32(unsigned u) {
    return (u & 0x80000000u) ? __uint_as_float(u ^ 0x80000000u) : __uint_as_float(~u);
}

// ---------------------------------------------------------------- edge scores + segment max
__global__ void gat_edge_score(const int* __restrict__ src,
                               const int* __restrict__ dst,
                               const float* __restrict__ el,
                               const float* __restrict__ er,
                               float* __restrict__ e,
                               unsigned* __restrict__ menc) {
    const int i = blockIdx.x * blockDim.x + threadIdx.x;
    if (i >= NEDGES) return;
    const int s = src[i], d = dst[i];
    float v = el[s] + er[d];
    v = (v >= 0.0f) ? v : NEG_SLOPE * v;          // leaky_relu
    e[i] = v;
    atomicMax(&menc[d], enc_f32(v));
}

// ---------------------------------------------------------------- exp + segment sum (denom)
__global__ void gat_edge_exp(const int* __restrict__ dst,
                             const unsigned* __restrict__ menc,
                             float* __restrict__ e,          // in: e, out: ex (in place)
                             float* __restrict__ denom) {
    const int i = blockIdx.x * blockDim.x + threadIdx.x;
    if (i >= NEDGES) return;
    const int d = dst[i];
    const float mx = dec_f32(menc[d]);
    const float ex = __expf(e[i] - mx);
    e[i] = ex;
    atomicAdd(&denom[d], ex);
}

// ---------------------------------------------------------------- weighted scatter: out += alpha * feat[src]
// One wave per edge; each lane handles a float4 chunk (coalesced 512B per edge).
__global__ void gat_scatter(const int* __restrict__ src,
                            const int* __restrict__ dst,
                            const float* __restrict__ ex,
                            const float* __restrict__ denom,
                            const float* __restrict__ feat,
                            float* __restrict__ out) {
    const int edge = blockIdx.x * 8 + (threadIdx.x >> 5);
    if (edge >= NEDGES) return;
    const int lane = threadIdx.x & 31;
    const int s = src[edge], d = dst[edge];
    const float alpha = ex[edge] / fmaxf(denom[d], 1e-9f);

    const float4 f = *(const float4*)(feat + (long)s * DFEAT + lane * 4);
    float* o = out + (long)d * DFEAT + lane * 4;
    atomicAdd(o + 0, alpha * f.x);
    atomicAdd(o + 1, alpha * f.y);
    atomicAdd(o + 2, alpha * f.z);
    atomicAdd(o + 3, alpha * f.w);
}

// ---------------------------------------------------------------- launcher
extern "C" void kernel_launch(void* const* d_in, const int* in_sizes, int n_in,
                              void* d_out, int out_size, void* d_ws, size_t ws_size,
                              hipStream_t stream) {
    const float* h      = (const float*)d_in[0];
    const int*   src    = (const int*)  d_in[1];
    const int*   dst    = (const int*)  d_in[2];
    const float* W      = (const float*)d_in[3];
    const float* attn_l = (const float*)d_in[4];
    const float* attn_r = (const float*)d_in[5];
    float* out = (float*)d_out;

    // workspace layout
    float*    feat  = (float*)d_ws;                         // N*128
    float*    el    = feat + (long)NNODES * DFEAT;          // N
    float*    er    = el + NNODES;                          // N
    unsigned* menc  = (unsigned*)(er + NNODES);             // N
    float*    denom = (float*)(menc + NNODES);              // N
    float*    e     = denom + NNODES;                       // E

    // zero: out, and [menc, denom) region (2N elements, bit-zero == encoded -inf / 0.0f)
    gat_zero_f32<<<4096, 256, 0, stream>>>(out, (long)NNODES * DFEAT);
    gat_zero_f32<<<256, 256, 0, stream>>>((float*)menc, 2L * NNODES);

    // feat = h @ W  (WMMA f32)
    const int nstrips = NNODES / 16;                        // 6250
    gat_gemm_wmma<<<(nstrips + 3) / 4, 128, 0, stream>>>(h, W, feat);

    // el/er
    gat_attn_logits<<<(NNODES + 3) / 4, 128, 0, stream>>>(feat, attn_l, attn_r, el, er);

    // edge scores + segment max
    gat_edge_score<<<(NEDGES + 255) / 256, 256, 0, stream>>>(src, dst, el, er, e, menc);

    // exp + denom
    gat_edge_exp<<<(NEDGES + 255) / 256, 256, 0, stream>>>(dst, menc, e, denom);

    // weighted scatter
    gat_scatter<<<(NEDGES + 7) / 8, 256, 0, stream>>>(src, dst, e, denom, feat, out);
}